// DiffusionGraphConv_16604343566383
// MI455X (gfx1250) — compile-verified
//
#include <hip/hip_runtime.h>
#include <hip/hip_bf16.h>

// GCN fused pipeline for MI455X (gfx1250, wave32).
//
//   z[d] = dinv[d]*( sum_{s->d} dinv[s]*x[s] + dinv[d]*x[d] )   (shared by both convs)
//   out  = relu(z@W1 + b1) + z@W2 + b2                          (fused dual WMMA GEMM)
//
// N=50000, E=800000, D=256.  ws: zpre[N*D] | deg[N] | dinv[N]

#define GCN_N 50000
#define GCN_E 800000
#define GCN_D 256

#define GCN_RT 5                    // row tiles per block (80 rows; 50000 = 625*80)
#define GCN_BROWS (16 * GCN_RT)
#define GCN_LDSTRIDE 260            // +4 float row pad -> conflict-free ds_load_b64

typedef __attribute__((ext_vector_type(2))) float v2f;
typedef __attribute__((ext_vector_type(4))) float v4f;
typedef __attribute__((ext_vector_type(8))) float v8f;

// ---------------- degree / normalization ----------------

__global__ void gcn_deg_init(float* __restrict__ deg) {
    int i = blockIdx.x * 256 + threadIdx.x;
    if (i < GCN_N) deg[i] = 1.0f;            // self loop
}

__global__ void gcn_deg_count(const long long* __restrict__ ei,
                              float* __restrict__ deg) {
    int e = blockIdx.x * 256 + threadIdx.x;
    if (e < GCN_E) {
        long long d = ei[GCN_E + e];         // dst row of edge_index
        unsafeAtomicAdd(&deg[d], 1.0f);      // global_atomic_add_f32
    }
}

__global__ void gcn_dinv(const float* __restrict__ deg,
                         float* __restrict__ dinv) {
    int i = blockIdx.x * 256 + threadIdx.x;
    if (i < GCN_N) {
        float dg = deg[i];
        dinv[i] = dg > 0.0f ? rsqrtf(dg) : 0.0f;
    }
}

// ---------------- aggregation of x into zpre ----------------

// zpre[d] = dinv[d]*x[d]   (self-loop term; also initializes ws). float4 lanes.
__global__ void gcn_zinit(const float* __restrict__ x,
                          const float* __restrict__ dinv,
                          float* __restrict__ zpre) {
    size_t i = (size_t)blockIdx.x * 256 + threadIdx.x;   // float4 index
    int row = (int)(i >> 6);                              // 64 float4 per row
    float w = dinv[row];
    v4f v = ((const v4f*)x)[i];
    v4f r = {v.x * w, v.y * w, v.z * w, v.w * w};
    ((v4f*)zpre)[i] = r;
}

// zpre[dst] += dinv[src]*x[src]   (4 edges per block, float4 per thread)
__global__ void gcn_scatter(const float* __restrict__ x,
                            const long long* __restrict__ ei,
                            const float* __restrict__ dinv,
                            float* __restrict__ zpre) {
    int t  = threadIdx.x;
    int e  = blockIdx.x * 4 + (t >> 6);
    int c4 = (t & 63) * 4;
    long long s = ei[e];
    long long d = ei[GCN_E + e];
    float w = dinv[s];
    v4f v = *(const v4f*)(x + (size_t)s * GCN_D + c4);
    float* dst = zpre + (size_t)d * GCN_D + c4;
    unsafeAtomicAdd(dst + 0, v.x * w);
    unsafeAtomicAdd(dst + 1, v.y * w);
    unsafeAtomicAdd(dst + 2, v.z * w);
    unsafeAtomicAdd(dst + 3, v.w * w);
}

// z = dinv[row] * zpre   (in place, float4 lanes)
__global__ void gcn_zscale(float* __restrict__ zpre,
                           const float* __restrict__ dinv) {
    size_t i = (size_t)blockIdx.x * 256 + threadIdx.x;
    int row = (int)(i >> 6);
    float w = dinv[row];
    v4f v = ((v4f*)zpre)[i];
    v4f r = {v.x * w, v.y * w, v.z * w, v.w * w};
    ((v4f*)zpre)[i] = r;
}

// ---------------- fused dual GEMM + bias + relu + add ----------------
// Block: 512 threads = 16 waves. Block bm owns rows [80*bm, 80*bm+80),
// wave w owns cols [16*w, 16*w+16) for all 5 row tiles. The 80x256 A-stripe
// is staged once in dynamic LDS; per k-step each wave loads one B fragment
// pair (W1,W2) and feeds it to 10 V_WMMA_F32_16X16X4_F32 ops (5 row tiles x
// 2 weights), amortizing B traffic 5x.

__global__ __launch_bounds__(512) void gcn_gemm_fused(
        const float* __restrict__ z,
        const float* __restrict__ W1, const float* __restrict__ b1,
        const float* __restrict__ W2, const float* __restrict__ b2,
        float* __restrict__ out) {
    extern __shared__ float As[];              // [GCN_BROWS][GCN_LDSTRIDE]

    int t = threadIdx.x;                       // 0..511
    int rowbase = blockIdx.x * GCN_BROWS;

    // cooperative load of 80x256 A-stripe: 5120 float4, 10 per thread
    const v4f* zsrc = (const v4f*)(z + (size_t)rowbase * GCN_D);
    for (int i = t; i < GCN_BROWS * 64; i += 512) {
        int r  = i >> 6;                       // 64 float4 per row
        int c4 = i & 63;
        *(v4f*)&As[r * GCN_LDSTRIDE + c4 * 4] = zsrc[i];
    }
    __syncthreads();

    int wave    = t >> 5;                      // 0..15 -> column tile
    int lane    = t & 31;
    int colbase = wave * 16;
    int n       = lane & 15;                   // B/C column within tile
    int m       = lane & 15;                   // A row within tile
    int koff    = (lane < 16) ? 0 : 2;         // f32 16x16x4 A/B lane split
    int col     = colbase + n;

    v8f c1[GCN_RT] = {};
    v8f c2[GCN_RT] = {};

    for (int k0 = 0; k0 < GCN_D; k0 += 4) {
        int kb = k0 + koff;
        // B fragments (K x N, row-major weights): lane<16 K=k0,k0+1; lane>=16 K=k0+2,k0+3
        v2f bA, bB;
        bA.x = W1[(size_t)kb * GCN_D + col];
        bA.y = W1[(size_t)(kb + 1) * GCN_D + col];
        bB.x = W2[(size_t)kb * GCN_D + col];
        bB.y = W2[(size_t)(kb + 1) * GCN_D + col];
#pragma unroll
        for (int rt = 0; rt < GCN_RT; ++rt) {
            v2f a = *(const v2f*)&As[(rt * 16 + m) * GCN_LDSTRIDE + kb];
            c1[rt] = __builtin_amdgcn_wmma_f32_16x16x4_f32(false, a, false, bA,
                                                           (short)0, c1[rt], false, false);
            c2[rt] = __builtin_amdgcn_wmma_f32_16x16x4_f32(false, a, false, bB,
                                                           (short)0, c2[rt], false, false);
        }
    }

    // epilogue: C/D layout -> VGPR r: lanes 0-15 = M=r, lanes 16-31 = M=r+8
    int mbase = (lane < 16) ? 0 : 8;
    float bb1 = b1[col];
    float bb2 = b2[col];
#pragma unroll
    for (int rt = 0; rt < GCN_RT; ++rt) {
#pragma unroll
        for (int r = 0; r < 8; ++r) {
            int row = rowbase + rt * 16 + mbase + r;
            float o1 = c1[rt][r] + bb1;
            o1 = o1 > 0.0f ? o1 : 0.0f;        // relu on conv1
            float o2 = c2[rt][r] + bb2;
            out[(size_t)row * GCN_D + col] = o1 + o2;
        }
    }
}

// ---------------- launcher ----------------

extern "C" void kernel_launch(void* const* d_in, const int* in_sizes, int n_in,
                              void* d_out, int out_size, void* d_ws, size_t ws_size,
                              hipStream_t stream) {
    const float*     x  = (const float*)d_in[0];
    const long long* ei = (const long long*)d_in[1];   // int64 edge_index [2,E]
    const float*     W1 = (const float*)d_in[2];
    const float*     b1 = (const float*)d_in[3];
    const float*     W2 = (const float*)d_in[4];
    const float*     b2 = (const float*)d_in[5];
    float* out = (float*)d_out;

    float* ws   = (float*)d_ws;
    float* zpre = ws;                                   // N*D floats
    float* deg  = ws + (size_t)GCN_N * GCN_D;           // N floats
    float* dinv = deg + GCN_N;                          // N floats

    // 1. degrees (self loop + dst counts) and rsqrt normalization
    gcn_deg_init <<<(GCN_N + 255) / 256, 256, 0, stream>>>(deg);
    gcn_deg_count<<<(GCN_E + 255) / 256, 256, 0, stream>>>(ei, deg);
    gcn_dinv     <<<(GCN_N + 255) / 256, 256, 0, stream>>>(deg, dinv);

    // 2. shared normalized aggregation of x (float4-wide)
    gcn_zinit  <<<(GCN_N * (GCN_D / 4)) / 256, 256, 0, stream>>>(x, dinv, zpre);
    gcn_scatter<<<GCN_E / 4, 256, 0, stream>>>(x, ei, dinv, zpre);
    gcn_zscale <<<(GCN_N * (GCN_D / 4)) / 256, 256, 0, stream>>>(zpre, dinv);

    // 3. fused dual GEMM (WMMA f32 16x16x4, 5 row tiles/wave) + bias + relu + add
    size_t lds_bytes = (size_t)GCN_BROWS * GCN_LDSTRIDE * sizeof(float);  // 83.2 KB
    gcn_gemm_fused<<<GCN_N / GCN_BROWS, 512, lds_bytes, stream>>>(
        zpre, W1, b1, W2, b2, out);
}